// LearnedPositionalEmbedding_4483945857469
// MI455X (gfx1250) — compile-verified
//
#include <hip/hip_runtime.h>
#include <cstdint>

typedef float v4f __attribute__((ext_vector_type(4)));

// Problem constants from the reference setup (deterministic).
#define BATCH  4
#define SEQ    4096
#define DMODEL 1024
#define MAXLEN 8192

#define TS   64   // s-extent of a tile
#define TD   64   // d-extent of a tile
#define LSTR 65   // LDS row stride in floats (64 + 1 pad -> conflict-free banking)

__global__ __launch_bounds__(256)
void posemb_tile_kernel(const float* __restrict__ W,
                        const float* __restrict__ bias,
                        float* __restrict__ out)
{
    // Tile stored TRANSPOSED: tile[s][d]  (transpose happens during the async copy)
    __shared__ float tile[TS * LSTR];

    const unsigned t       = threadIdx.x;
    const unsigned tiles_d = DMODEL / TD;                 // 16
    const unsigned s0      = (blockIdx.x / tiles_d) * TS;
    const unsigned d0      = (blockIdx.x % tiles_d) * TD;

    // Low 32 bits of a generic pointer into LDS == LDS byte offset (ISA 10.2).
    const unsigned lds_base = (unsigned)(uintptr_t)&tile[0];

    // ---- Phase 1: W[d0:d0+64, s0:s0+64] -> LDS, transposed, via async DMA ----
    // Lanes consecutive in s: each wave instruction reads 128B contiguous from
    // one W row. LDS dest bank = (s + d) mod 64 -> conflict-free.
    {
        const unsigned s_local = t & 63u;   // 0..63
        const unsigned d_base  = t >> 6;    // 0..3
#pragma unroll
        for (unsigned i = 0; i < 16; ++i) {
            const unsigned d_local = d_base * 16u + i;       // 0..63
            const unsigned goff =
                ((d0 + d_local) * (unsigned)MAXLEN + s0 + s_local) * 4u;
            const unsigned loff =
                lds_base + (s_local * (unsigned)LSTR + d_local) * 4u;
            // GVS mode: mem = SGPR64(W) + VGPR32(goff); LDS addr from VGPR(loff)
            asm volatile("global_load_async_to_lds_b32 %0, %1, %2"
                         :
                         : "v"(loff), "v"(goff), "s"(W)
                         : "memory");
        }
        asm volatile("s_wait_asynccnt 0" ::: "memory");
    }
    __syncthreads();

    // ---- Phase 2: bias add + broadcast to all batches, coalesced NT stores ----
    const unsigned c     = t & 15u;  // which float4 along d
    const unsigned s_sub = t >> 4;   // 0..15
    const v4f bv = *(const v4f*)(bias + d0 + c * 4u);   // 16B aligned

#pragma unroll
    for (unsigned j = 0; j < 4; ++j) {
        const unsigned s_local = s_sub + 16u * j;
        // Contiguous-in-d reads from the transposed tile; stride-65 rows make
        // these scalar ds_load_b32, banks (s + 4c + k) mod 64: conflict-free.
        v4f v;
        v.x = tile[s_local * LSTR + c * 4u + 0] + bv.x;
        v.y = tile[s_local * LSTR + c * 4u + 1] + bv.y;
        v.z = tile[s_local * LSTR + c * 4u + 2] + bv.z;
        v.w = tile[s_local * LSTR + c * 4u + 3] + bv.w;

        const size_t rowbase =
            (size_t)(s0 + s_local) * (size_t)DMODEL + d0 + c * 4u;
#pragma unroll
        for (unsigned b = 0; b < BATCH; ++b) {
            v4f* dst = (v4f*)(out + (size_t)b * (size_t)SEQ * (size_t)DMODEL
                                  + rowbase);
            __builtin_nontemporal_store(v, dst);  // 16B NT store, coalesced
        }
    }
}

extern "C" void kernel_launch(void* const* d_in, const int* /*in_sizes*/,
                              int /*n_in*/, void* d_out, int /*out_size*/,
                              void* /*d_ws*/, size_t /*ws_size*/,
                              hipStream_t stream)
{
    // d_in[0] = x [4,4096] (shape-only, unused)
    const float* W    = (const float*)d_in[1];  // [1024, 8192]
    const float* bias = (const float*)d_in[2];  // [1024]
    float* out = (float*)d_out;                 // [4, 4096, 1024]

    dim3 grid((SEQ / TS) * (DMODEL / TD));      // 64 * 16 = 1024 tiles
    dim3 block(256);
    posemb_tile_kernel<<<grid, block, 0, stream>>>(W, bias, out);
}